// Attention_1151051235741
// MI455X (gfx1250) — compile-verified
//
#include <hip/hip_runtime.h>

typedef __attribute__((ext_vector_type(2))) float v2f;
typedef __attribute__((ext_vector_type(8))) float v8f;

#define WMMA_F32(a, b, c) \
  __builtin_amdgcn_wmma_f32_16x16x4_f32(false, (a), false, (b), (short)0, (c), false, false)

// ---------------- fragment helpers (ISA 7.12.2 layouts, wave32) ----------------
// A (16x4 f32): element (m,k): lane = (k/2)*16 + m, vgpr = k%2
__device__ __forceinline__ v2f ldA(const float* A, int lda, int row0, int k0, int lane) {
  int m = lane & 15, hh = lane >> 4;
  const float* p = A + (size_t)(row0 + m) * lda + (k0 + 2 * hh);
  v2f r; r.x = p[0]; r.y = p[1]; return r;
}
// B (4x16 f32): element (k,n): lane = (k/2)*16 + n, vgpr = k%2
__device__ __forceinline__ v2f ldB(const float* B, int ldb, int k0, int col0, int lane) {
  int n = lane & 15, hh = lane >> 4;
  const float* p = B + (size_t)(k0 + 2 * hh) * ldb + (col0 + n);
  v2f r; r.x = p[0]; r.y = p[ldb]; return r;
}
// B = Bt^T where Bt is row-major (n x k): B[k][n] = Bt[n][k]
__device__ __forceinline__ v2f ldBT(const float* Bt, int ldbt, int k0, int col0, int lane) {
  int n = lane & 15, hh = lane >> 4;
  const float* p = Bt + (size_t)(col0 + n) * ldbt + (k0 + 2 * hh);
  v2f r; r.x = p[0]; r.y = p[1]; return r;
}
// C/D (16x16 f32): vgpr r, lane: row = 8*(lane>>4)+r, col = lane&15
__device__ __forceinline__ void stC(float* C, int ldc, int row0, int col0, int lane, v8f acc) {
  int n = lane & 15, mh = (lane >> 4) * 8;
#pragma unroll
  for (int r = 0; r < 8; ++r) C[(size_t)(row0 + mh + r) * ldc + col0 + n] = acc[r];
}
__device__ __forceinline__ void ldCtile(const float* C, int ldc, int row0, int col0, int lane, float out[8]) {
  int n = lane & 15, mh = (lane >> 4) * 8;
#pragma unroll
  for (int r = 0; r < 8; ++r) out[r] = C[(size_t)(row0 + mh + r) * ldc + col0 + n];
}

// ---------------- sizes ----------------
#define B_  4
#define N_  2048
#define C_  512
#define H_  8
#define D_  64
#define M_  64
#define BH_ 32
#define BN_ 8192

// ---------------- K1: qkv = x @ w_qkv, scatter to head-major q(scaled)/k/v ----------------
__global__ void k_qkv(const float* __restrict__ x, const float* __restrict__ w,
                      float* __restrict__ qh, float* __restrict__ kh, float* __restrict__ vh) {
  const int lane = threadIdx.x & 31, wave = threadIdx.x >> 5;
  const int ntile = blockIdx.x * 8 + wave;   // 96 tiles of 1536
  const int m0 = blockIdx.y * 16;            // 512 tiles of 8192
  const int n0 = ntile * 16;
  v8f acc = {};
  for (int k0 = 0; k0 < C_; k0 += 4) {
    v2f a = ldA(x, C_, m0, k0, lane);
    v2f b = ldB(w, 3 * C_, k0, n0, lane);
    acc = WMMA_F32(a, b, acc);
  }
  const int n = lane & 15, mh = (lane >> 4) * 8;
#pragma unroll
  for (int r = 0; r < 8; ++r) {
    int row = m0 + mh + r;
    int bb = row >> 11, seq = row & (N_ - 1);
    int col = n0 + n;
    int part = col >> 9, cc = col & (C_ - 1);
    int head = cc >> 6, dd = cc & 63;
    size_t o = (((size_t)(bb * H_ + head)) * N_ + seq) * D_ + dd;
    float val = acc[r];
    if (part == 0)      qh[o] = val * 0.125f;   // scale = d^-0.5
    else if (part == 1) kh[o] = val;
    else                vh[o] = val;
  }
}

// ---------------- K2: landmark means over contiguous segments of 32 ----------------
__global__ void k_land(const float* __restrict__ qh, const float* __restrict__ kh,
                       float* __restrict__ ql, float* __restrict__ kl) {
  int idx = blockIdx.x * blockDim.x + threadIdx.x;  // BH*M*D = 131072
  int dd = idx & 63, i = (idx >> 6) & 63, bh = idx >> 12;
  const float* qp = qh + ((size_t)bh * N_ + i * 32) * D_ + dd;
  const float* kp = kh + ((size_t)bh * N_ + i * 32) * D_ + dd;
  float sq = 0.f, sk = 0.f;
  for (int j = 0; j < 32; ++j) { sq += qp[j * D_]; sk += kp[j * D_]; }
  ql[idx] = sq * (1.0f / 32.0f);
  kl[idx] = sk * (1.0f / 32.0f);
}

// ---------------- K3: attn1 = softmax(q @ kl^T) rows of 64, fused ----------------
__global__ void k_attn1(const float* __restrict__ qh, const float* __restrict__ kl,
                        float* __restrict__ attn1) {
  __shared__ float s[16 * 64];
  const int lane = threadIdx.x & 31, wave = threadIdx.x >> 5;  // 4 waves
  const int bh = blockIdx.y, m0 = blockIdx.x * 16;
  const float* A = qh + (size_t)bh * N_ * D_;
  const float* KL = kl + (size_t)bh * M_ * D_;
  v8f acc = {};
  for (int k0 = 0; k0 < D_; k0 += 4) {
    v2f a = ldA(A, D_, m0, k0, lane);
    v2f b = ldBT(KL, D_, k0, wave * 16, lane);
    acc = WMMA_F32(a, b, acc);
  }
  stC(s, 64, 0, wave * 16, lane, acc);
  __syncthreads();
  if (threadIdx.x < 16) {
    int r = threadIdx.x;
    float mx = -3.4e38f;
    for (int j = 0; j < 64; ++j) mx = fmaxf(mx, s[r * 64 + j]);
    float sum = 0.f;
    for (int j = 0; j < 64; ++j) { float e = __expf(s[r * 64 + j] - mx); s[r * 64 + j] = e; sum += e; }
    float inv = 1.0f / sum;
    float* o = attn1 + ((size_t)bh * N_ + m0 + r) * M_;
    for (int j = 0; j < 64; ++j) o[j] = s[r * 64 + j] * inv;
  }
}

// ---------------- K4: attn2 = softmax(ql @ kl^T), 64x64 per bh ----------------
__global__ void k_attn2(const float* __restrict__ ql, const float* __restrict__ kl,
                        float* __restrict__ attn2) {
  __shared__ float s[64 * 64];
  const int lane = threadIdx.x & 31, wave = threadIdx.x >> 5;  // 16 waves
  const int bh = blockIdx.x;
  const float* A = ql + (size_t)bh * M_ * D_;
  const float* KL = kl + (size_t)bh * M_ * D_;
  int mt = (wave >> 2) * 16, nt = (wave & 3) * 16;
  v8f acc = {};
  for (int k0 = 0; k0 < D_; k0 += 4) {
    v2f a = ldA(A, D_, mt, k0, lane);
    v2f b = ldBT(KL, D_, k0, nt, lane);
    acc = WMMA_F32(a, b, acc);
  }
  stC(s, 64, mt, nt, lane, acc);
  __syncthreads();
  if (threadIdx.x < 64) {
    int r = threadIdx.x;
    float mx = -3.4e38f;
    for (int j = 0; j < 64; ++j) mx = fmaxf(mx, s[r * 64 + j]);
    float sum = 0.f;
    for (int j = 0; j < 64; ++j) { float e = __expf(s[r * 64 + j] - mx); s[r * 64 + j] = e; sum += e; }
    float inv = 1.0f / sum;
    float* o = attn2 + (size_t)bh * M_ * M_ + (size_t)r * M_;
    for (int j = 0; j < 64; ++j) o[j] = s[r * 64 + j] * inv;
  }
}

// ---------------- K5a: sim3 = ql @ k^T (64 x 2048 per bh), raw ----------------
__global__ void k_sim3(const float* __restrict__ ql, const float* __restrict__ kh,
                       float* __restrict__ attn3) {
  const int lane = threadIdx.x & 31, wave = threadIdx.x >> 5;  // 8 waves
  const int bh = blockIdx.y;
  const int ntile = blockIdx.x * 8 + wave;  // 128 col tiles
  const float* A = ql + (size_t)bh * M_ * D_;
  const float* K = kh + (size_t)bh * N_ * D_;
  v8f acc[4] = {{}, {}, {}, {}};
  for (int k0 = 0; k0 < D_; k0 += 4) {
    v2f b = ldBT(K, D_, k0, ntile * 16, lane);
#pragma unroll
    for (int t = 0; t < 4; ++t) {
      v2f a = ldA(A, D_, t * 16, k0, lane);
      acc[t] = WMMA_F32(a, b, acc[t]);
    }
  }
  float* Cp = attn3 + (size_t)bh * M_ * N_;
#pragma unroll
  for (int t = 0; t < 4; ++t) stC(Cp, N_, t * 16, ntile * 16, lane, acc[t]);
}

// ---------------- K5b: in-place row softmax, row length = rowlen ----------------
__global__ void k_softmax_row(float* __restrict__ mat, int rowlen) {
  __shared__ float red[256];
  float* p = mat + (size_t)blockIdx.x * rowlen;
  int t = threadIdx.x;
  float mx = -3.4e38f;
  for (int j = t; j < rowlen; j += 256) mx = fmaxf(mx, p[j]);
  red[t] = mx; __syncthreads();
  for (int s = 128; s > 0; s >>= 1) { if (t < s) red[t] = fmaxf(red[t], red[t + s]); __syncthreads(); }
  mx = red[0]; __syncthreads();
  float sum = 0.f;
  for (int j = t; j < rowlen; j += 256) { float e = __expf(p[j] - mx); p[j] = e; sum += e; }
  red[t] = sum; __syncthreads();
  for (int s = 128; s > 0; s >>= 1) { if (t < s) red[t] += red[t + s]; __syncthreads(); }
  float inv = 1.0f / red[0];
  for (int j = t; j < rowlen; j += 256) p[j] *= inv;
}

// ---------------- K6a: global max of abs row/col sums of attn2 ----------------
__global__ void k_zero2(unsigned* p) { if (threadIdx.x < 2) p[threadIdx.x] = 0u; }

__global__ void k_norm(const float* __restrict__ attn2, unsigned* __restrict__ norms) {
  __shared__ float rc[64], rr[64];
  int bh = blockIdx.x, j = threadIdx.x;  // 64 threads
  const float* a = attn2 + (size_t)bh * M_ * M_;
  float cs = 0.f, rs = 0.f;
  for (int i = 0; i < 64; ++i) { cs += fabsf(a[j * 64 + i]); rs += fabsf(a[i * 64 + j]); }
  rc[j] = cs; rr[j] = rs; __syncthreads();
  for (int s = 32; s > 0; s >>= 1) {
    if (j < s) { rc[j] = fmaxf(rc[j], rc[j + s]); rr[j] = fmaxf(rr[j], rr[j + s]); }
    __syncthreads();
  }
  if (j == 0) {  // sums are positive -> bit compare is monotone
    atomicMax(&norms[0], __float_as_uint(rc[0]));
    atomicMax(&norms[1], __float_as_uint(rr[0]));
  }
}

// ---------------- K6b: Newton-Schulz pinv, 6 iters, all in LDS ----------------
__global__ void k_pinv(const float* __restrict__ attn2, const unsigned* __restrict__ norms,
                       float* __restrict__ zout) {
  __shared__ float Z[64 * 64], B1[64 * 64], B2[64 * 64];
  const int lane = threadIdx.x & 31, wave = threadIdx.x >> 5;  // 16 waves
  const int bh = blockIdx.x;
  const float* A = attn2 + (size_t)bh * M_ * M_;
  const float inv = 1.0f / (__uint_as_float(norms[0]) * __uint_as_float(norms[1]));
  for (int idx = threadIdx.x; idx < 4096; idx += 512) {
    int i = idx >> 6, jj = idx & 63;
    Z[idx] = A[jj * 64 + i] * inv;   // Z = A^T / (maxcol*maxrow)
  }
  __syncthreads();
  const int mt = (wave >> 2) * 16, nt = (wave & 3) * 16;
  for (int iter = 0; iter < 6; ++iter) {
    // B1 = A @ Z
    v8f acc = {};
    for (int k0 = 0; k0 < 64; k0 += 4) {
      v2f a = ldA(A, 64, mt, k0, lane);
      v2f b = ldB(Z, 64, k0, nt, lane);
      acc = WMMA_F32(a, b, acc);
    }
    stC(B1, 64, mt, nt, lane, acc);
    __syncthreads();
    // B2 = 7*B1 - B1@B1
    acc = (v8f){};
    for (int k0 = 0; k0 < 64; k0 += 4) {
      v2f a = ldA(B1, 64, mt, k0, lane);
      v2f b = ldB(B1, 64, k0, nt, lane);
      acc = WMMA_F32(a, b, acc);
    }
    float c[8];
    ldCtile(B1, 64, mt, nt, lane, c);
#pragma unroll
    for (int r = 0; r < 8; ++r) acc[r] = 7.0f * c[r] - acc[r];
    stC(B2, 64, mt, nt, lane, acc);
    __syncthreads();
    // B1' = 15*B1 - B1@B2  (stage in regs, overwrite after barrier)
    acc = (v8f){};
    for (int k0 = 0; k0 < 64; k0 += 4) {
      v2f a = ldA(B1, 64, mt, k0, lane);
      v2f b = ldB(B2, 64, k0, nt, lane);
      acc = WMMA_F32(a, b, acc);
    }
    ldCtile(B1, 64, mt, nt, lane, c);
#pragma unroll
    for (int r = 0; r < 8; ++r) acc[r] = 15.0f * c[r] - acc[r];
    __syncthreads();
    stC(B1, 64, mt, nt, lane, acc);
    __syncthreads();
    // Z' = 0.25 * (13*Z - Z@B1)
    acc = (v8f){};
    for (int k0 = 0; k0 < 64; k0 += 4) {
      v2f a = ldA(Z, 64, mt, k0, lane);
      v2f b = ldB(B1, 64, k0, nt, lane);
      acc = WMMA_F32(a, b, acc);
    }
    ldCtile(Z, 64, mt, nt, lane, c);
#pragma unroll
    for (int r = 0; r < 8; ++r) acc[r] = 0.25f * (13.0f * c[r] - acc[r]);
    __syncthreads();
    stC(Z, 64, mt, nt, lane, acc);
    __syncthreads();
  }
  float* o = zout + (size_t)bh * M_ * M_;
  for (int idx = threadIdx.x; idx < 4096; idx += 512) o[idx] = Z[idx];
}

// ---------------- K7/K9: C(2048x64) = A(2048x64) @ B(64x64) per bh ----------------
__global__ void k_gemm_n64(const float* __restrict__ Abase, const float* __restrict__ Bbase,
                           float* __restrict__ Cbase) {
  const int lane = threadIdx.x & 31, wave = threadIdx.x >> 5;  // 8 waves
  const int bh = blockIdx.y;
  int id = blockIdx.x * 8 + wave;  // 512 wave-tiles per bh
  int mt = id >> 2, nt = id & 3;
  const float* A = Abase + (size_t)bh * N_ * 64;
  const float* B = Bbase + (size_t)bh * 64 * 64;
  v8f acc = {};
  for (int k0 = 0; k0 < 64; k0 += 4) {
    v2f a = ldA(A, 64, mt * 16, k0, lane);
    v2f b = ldB(B, 64, k0, nt * 16, lane);
    acc = WMMA_F32(a, b, acc);
  }
  stC(Cbase + (size_t)bh * N_ * 64, 64, mt * 16, nt * 16, lane, acc);
}

// ---------------- K8: a3v = attn3(64x2048) @ v(2048x64) per bh ----------------
__global__ void k_a3v(const float* __restrict__ attn3, const float* __restrict__ vh,
                      float* __restrict__ a3v) {
  const int lane = threadIdx.x & 31, wave = threadIdx.x >> 5;  // 16 waves
  const int bh = blockIdx.x;
  const float* A = attn3 + (size_t)bh * M_ * N_;
  const float* B = vh + (size_t)bh * N_ * D_;
  int mt = (wave >> 2) * 16, nt = (wave & 3) * 16;
  v8f acc = {};
  for (int k0 = 0; k0 < N_; k0 += 4) {
    v2f a = ldA(A, N_, mt, k0, lane);
    v2f b = ldB(B, D_, k0, nt, lane);
    acc = WMMA_F32(a, b, acc);
  }
  stC(a3v + (size_t)bh * M_ * D_, D_, mt, nt, lane, acc);
}

// ---------------- K10: attn = tmp(2048x64) @ attn3(64x2048), NT stores ----------------
__global__ void k_attn_out(const float* __restrict__ tmp, const float* __restrict__ attn3,
                           float* __restrict__ attn) {
  const int lane = threadIdx.x & 31, wave = threadIdx.x >> 5;  // 8 waves
  const int bh = blockIdx.y;
  int id = blockIdx.x * 8 + wave;       // 4096 wave-strips per bh
  int m0 = (id >> 5) * 16, n0 = (id & 31) * 64;
  const float* A = tmp + (size_t)bh * N_ * M_;
  const float* B = attn3 + (size_t)bh * M_ * N_;
  v8f acc[4] = {{}, {}, {}, {}};
  for (int k0 = 0; k0 < 64; k0 += 4) {
    v2f a = ldA(A, M_, m0, k0, lane);   // A-frag reused across 4 N-tiles
#pragma unroll
    for (int t = 0; t < 4; ++t) {
      v2f b = ldB(B, N_, k0, n0 + t * 16, lane);
      acc[t] = WMMA_F32(a, b, acc[t]);
    }
  }
  float* C = attn + (size_t)bh * N_ * N_;
  int n = lane & 15, mh = (lane >> 4) * 8;
#pragma unroll
  for (int t = 0; t < 4; ++t)
#pragma unroll
    for (int r = 0; r < 8; ++r)
      __builtin_nontemporal_store(acc[t][r], &C[(size_t)(m0 + mh + r) * N_ + n0 + t * 16 + n]);
}

// ---------------- K11: out = gather(outh) @ w_proj + b_proj ----------------
__global__ void k_proj(const float* __restrict__ outh, const float* __restrict__ w,
                       const float* __restrict__ bias, float* __restrict__ out) {
  const int lane = threadIdx.x & 31, wave = threadIdx.x >> 5;  // 8 waves
  int id = blockIdx.x * 8 + wave;  // 16384 wave-tiles
  int m0 = (id >> 5) * 16, n0 = (id & 31) * 16;
  int mm = lane & 15, hh = lane >> 4;
  v8f acc = {};
  for (int k0 = 0; k0 < C_; k0 += 4) {
    int kk = k0 + 2 * hh;
    int row = m0 + mm;
    int bb = row >> 11, seq = row & (N_ - 1);
    // A[b*2048+seq][c2] = outh[(b*8 + c2/64)][seq][c2%64]; kk even -> pair stays in-head
    const float* ap = outh + (((size_t)(bb * H_ + (kk >> 6)) * N_ + seq) * D_ + (kk & 63));
    v2f a; a.x = ap[0]; a.y = ap[1];
    v2f b = ldB(w, C_, k0, n0, lane);
    acc = WMMA_F32(a, b, acc);
  }
  int n = lane & 15, mh = (lane >> 4) * 8;
#pragma unroll
  for (int r = 0; r < 8; ++r)
    out[(size_t)(m0 + mh + r) * C_ + n0 + n] = acc[r] + bias[n0 + n];
}

// ---------------- launch ----------------
extern "C" void kernel_launch(void* const* d_in, const int* in_sizes, int n_in,
                              void* d_out, int out_size, void* d_ws, size_t ws_size,
                              hipStream_t stream) {
  (void)in_sizes; (void)n_in; (void)out_size; (void)ws_size;
  const float* x      = (const float*)d_in[0];
  const float* w_qkv  = (const float*)d_in[1];
  const float* w_proj = (const float*)d_in[2];
  const float* b_proj = (const float*)d_in[3];

  float* out  = (float*)d_out;                 // 4*2048*512
  float* attn = (float*)d_out + 4194304;       // 4*8*2048*2048

  float* ws = (float*)d_ws;                    // offsets in floats (~120 MB total)
  float* qh    = ws;                           // 4,194,304
  float* kh    = ws + 4194304;
  float* vh    = ws + 8388608;
  float* qland = ws + 12582912;                // 131,072
  float* kland = ws + 12713984;
  float* attn1 = ws + 12845056;                // 4,194,304
  float* attn3 = ws + 17039360;                // 4,194,304
  float* attn2 = ws + 21233664;                // 131,072
  float* zmat  = ws + 21364736;                // 131,072
  float* tmp   = ws + 21495808;                // 4,194,304
  float* a3v   = ws + 25690112;                // 131,072
  float* outh  = ws + 25821184;                // 4,194,304
  unsigned* norms = (unsigned*)(ws + 30015488);

  k_zero2<<<1, 32, 0, stream>>>(norms);
  k_qkv<<<dim3(12, 512), 256, 0, stream>>>(x, w_qkv, qh, kh, vh);
  k_land<<<512, 256, 0, stream>>>(qh, kh, qland, kland);
  k_attn1<<<dim3(128, 32), 128, 0, stream>>>(qh, kland, attn1);
  k_attn2<<<32, 512, 0, stream>>>(qland, kland, attn2);
  k_sim3<<<dim3(16, 32), 256, 0, stream>>>(qland, kh, attn3);
  k_softmax_row<<<2048, 256, 0, stream>>>(attn3, 2048);
  k_norm<<<32, 64, 0, stream>>>(attn2, norms);
  k_pinv<<<32, 512, 0, stream>>>(attn2, norms, zmat);
  k_gemm_n64<<<dim3(64, 32), 256, 0, stream>>>(attn1, zmat, tmp);   // tmp = attn1 @ pinv
  k_a3v<<<32, 512, 0, stream>>>(attn3, vh, a3v);                    // attn3 @ v
  k_gemm_n64<<<dim3(64, 32), 256, 0, stream>>>(tmp, a3v, outh);     // out heads
  k_attn_out<<<dim3(512, 32), 256, 0, stream>>>(tmp, attn3, attn);  // big NT-stored output
  k_proj<<<2048, 256, 0, stream>>>(outh, w_proj, b_proj, out);
}